// StreamingPCENTransform_63788854280996
// MI455X (gfx1250) — compile-verified
//
#include <hip/hip_runtime.h>
#include <hip/hip_bf16.h>

// ---------------- problem constants ----------------
#define N_FFT     1024
#define HOP       256
#define N_MELS    128
#define L_IN      160000
#define PAD       512          // N_FFT/2
#define N_FRAMES  626          // (L_IN + 2*PAD - N_FFT)/HOP + 1
#define BATCH     32
#define FRAMES_TOTAL (BATCH * N_FRAMES)   // 20032 (multiple of 16)
#define N_FREQ    513          // N_FFT/2 + 1
#define KP        516          // K padded to multiple of 4 for WMMA f32 16x16x4
#define S_COEF    0.025f
#define A_COEF    0.975f       // 1 - S
#define ALPHA_C   0.98f
#define EPS_C     1e-6f

// scan chunking
#define N_CHUNK   16
#define CHUNK_LEN 40           // 15*40 + 26 = 626

typedef __attribute__((ext_vector_type(2))) float v2f;
typedef __attribute__((ext_vector_type(8))) float v8f;

// ---------------------------------------------------------------------------
// Kernel 1: framing (reflect pad) + 1024-pt radix-2 FFT in LDS + |.| -> spec
// One workgroup (256 threads) per frame.  Twiddles from a 512-entry LDS table
// (one sincos per entry instead of one per butterfly).  spec row stride = KP.
// ---------------------------------------------------------------------------
__global__ __launch_bounds__(256) void fft_mag_kernel(
    const float* __restrict__ x, float* __restrict__ spec)
{
    __shared__ float re[N_FFT];
    __shared__ float im[N_FFT];
    __shared__ float twr[N_FFT / 2];
    __shared__ float twi[N_FFT / 2];

    const int frame = blockIdx.x;            // 0 .. FRAMES_TOTAL-1
    const int b = frame / N_FRAMES;
    const int t = frame - b * N_FRAMES;
    const float* xb = x + (size_t)b * L_IN;
    const int tid = threadIdx.x;

    // twiddle table: W_N^k = exp(-2*pi*i*k/N), k = 0..511
    #pragma unroll
    for (int k = tid; k < N_FFT / 2; k += 256) {
        float sn, cs;
        sincosf((-6.283185307179586f / (float)N_FFT) * (float)k, &sn, &cs);
        twr[k] = cs;
        twi[k] = sn;
    }

    // load with reflect padding, bit-reversed placement (10-bit reverse)
    #pragma unroll
    for (int j = tid; j < N_FFT; j += 256) {
        int si = t * HOP + j - PAD;
        if (si < 0)        si = -si;                 // reflect (no edge dup)
        if (si > L_IN - 1) si = 2 * (L_IN - 1) - si;
        int r = (int)(__brev((unsigned)j) >> 22);    // 10-bit bit reverse
        re[r] = xb[si];
        im[r] = 0.0f;
    }
    __syncthreads();

    // iterative DIT radix-2, 10 stages, 512 butterflies/stage
    int tstep = N_FFT / 2;                   // twiddle index stride = 512/half
    for (int half = 1; half < N_FFT; half <<= 1, tstep >>= 1) {
        #pragma unroll
        for (int i = tid; i < (N_FFT / 2); i += 256) {
            const int j   = i & (half - 1);
            const int pos = (i << 1) - j;            // blk*len + j
            const int ti  = j * tstep;
            const float cs = twr[ti], sn = twi[ti];
            const float ur = re[pos],        ui = im[pos];
            const float vr = re[pos + half], vi = im[pos + half];
            const float tr = vr * cs - vi * sn;
            const float tg = vr * sn + vi * cs;
            re[pos]        = ur + tr;  im[pos]        = ui + tg;
            re[pos + half] = ur - tr;  im[pos + half] = ui - tg;
        }
        __syncthreads();
    }

    // magnitudes for bins 0..512, zero pad 513..515
    float* srow = spec + (size_t)frame * KP;
    for (int f = tid; f < KP; f += 256) {
        float m = 0.0f;
        if (f < N_FREQ) m = sqrtf(re[f] * re[f] + im[f] * im[f]);
        srow[f] = m;
    }
}

// ---------------------------------------------------------------------------
// Kernel 2: mel = spec(20032 x 516, zero-padded K) @ fb(513 x 128)
// FP32 WMMA 16x16x4.  One wave -> one 16x16 output tile; 8 waves/block.
// Main K loop (k = 0..508) is branch-free; final step (k = 512) peeled with a
// lane-level select (no EXEC manipulation -> WMMA-legal, no divergence).
// ---------------------------------------------------------------------------
__global__ __launch_bounds__(256) void mel_gemm_kernel(
    const float* __restrict__ spec, const float* __restrict__ fb,
    float* __restrict__ mel)
{
    const int wave = threadIdx.x >> 5;        // 0..7
    const int lane = threadIdx.x & 31;
    const int lm   = lane & 15;               // 0..15
    const int lh   = lane >> 4;               // 0 or 1
    const int row0 = blockIdx.x * 16;         // frame-row base
    const int n0   = wave * 16;               // mel-channel base

    v8f acc = {};
    const float* arow = spec + (size_t)(row0 + lm) * KP;
    const float* bcol = fb + n0 + lm;

    // main loop: ka = k + 2*lh <= 510, ka+1 <= 511 < 513 -> always in-bounds
    for (int k = 0; k < 512; k += 4) {
        const int ka = k + 2 * lh;            // even -> 8B aligned
        v2f a = *(const v2f*)(arow + ka);
        v2f bfr;
        bfr.x = bcol[(size_t)ka       * N_MELS];
        bfr.y = bcol[(size_t)(ka + 1) * N_MELS];
        acc = __builtin_amdgcn_wmma_f32_16x16x4_f32(
            false, a, false, bfr, (short)0, acc, false, false);
    }

    // tail: k = 512.  Only K=512 is real (K=513..515 are zero).
    {
        const int ka = 512 + 2 * lh;          // 512 or 514; spec is zero-padded
        v2f a = *(const v2f*)(arow + ka);
        const float blast = bcol[(size_t)512 * N_MELS];  // valid for all lanes
        v2f bfr;
        bfr.x = (lh == 0) ? blast : 0.0f;     // cndmask, no branch
        bfr.y = 0.0f;                         // K=513/515 rows don't exist
        acc = __builtin_amdgcn_wmma_f32_16x16x4_f32(
            false, a, false, bfr, (short)0, acc, false, false);
    }

    const int mbase = row0 + lh * 8;
    #pragma unroll
    for (int r = 0; r < 8; ++r)
        mel[(size_t)(mbase + r) * N_MELS + n0 + lm] = acc[r];
}

// ---------------------------------------------------------------------------
// EMA scan, block-parallel over time:  M_t = a*M_{t-1} + s*v_t,  M_0 = v_0.
// Pass A: per-chunk local scan endpoints (zero initial; chunk 0 exact).
// Pass B: serial combine over the 16 chunk endpoints (closed form).
// Pass C: re-run each chunk from its corrected initial state, emit PCEN.
// ---------------------------------------------------------------------------
__global__ __launch_bounds__(128) void scan_chunk_kernel(
    const float* __restrict__ mel, float* __restrict__ E)
{
    const int blk = blockIdx.x;               // 0 .. BATCH*N_CHUNK-1
    const int b = blk / N_CHUNK;
    const int c = blk - b * N_CHUNK;
    const int m = threadIdx.x;
    const int t0 = c * CHUNK_LEN;
    const int t1 = min(t0 + CHUNK_LEN, N_FRAMES);

    const size_t base = ((size_t)b * N_FRAMES) * N_MELS + m;
    float M;
    int t;
    if (c == 0) { M = mel[base]; t = 1; }
    else        { M = 0.0f;      t = t0; }
    for (; t < t1; ++t)
        M = A_COEF * M + S_COEF * mel[base + (size_t)t * N_MELS];
    E[(size_t)blk * N_MELS + m] = M;
}

__global__ __launch_bounds__(128) void scan_combine_kernel(float* __restrict__ E)
{
    const int b = blockIdx.x;                 // 0 .. BATCH-1
    const int m = threadIdx.x;
    const float decay_full = powf(A_COEF, (float)CHUNK_LEN);          // chunks 1..14
    const float decay_last = powf(A_COEF, (float)(N_FRAMES - (N_CHUNK - 1) * CHUNK_LEN));

    float prev = E[((size_t)b * N_CHUNK) * N_MELS + m];
    for (int c = 1; c < N_CHUNK; ++c) {
        const float d = (c == N_CHUNK - 1) ? decay_last : decay_full;
        const size_t idx = ((size_t)b * N_CHUNK + c) * N_MELS + m;
        prev = E[idx] + d * prev;
        E[idx] = prev;                        // E now holds true boundary M
    }
}

__global__ __launch_bounds__(128) void pcen_emit_kernel(
    const float* __restrict__ mel, const float* __restrict__ E,
    float* __restrict__ out)
{
    const int blk = blockIdx.x;               // 0 .. BATCH*N_CHUNK-1
    const int b = blk / N_CHUNK;
    const int c = blk - b * N_CHUNK;
    const int m = threadIdx.x;
    int t0 = c * CHUNK_LEN;
    const int t1 = min(t0 + CHUNK_LEN, N_FRAMES);
    const float sqrt_delta = 1.4142135623730951f;   // DELTA^R (DELTA=2, R=0.5)

    const size_t base = ((size_t)b * N_FRAMES) * N_MELS + m;
    float M;
    if (c == 0) {
        const float v = mel[base];
        M = v;
        out[base] = sqrtf(v / powf(M + EPS_C, ALPHA_C) + 2.0f) - sqrt_delta;
        t0 = 1;
    } else {
        M = E[((size_t)b * N_CHUNK + (c - 1)) * N_MELS + m];
    }
    for (int t = t0; t < t1; ++t) {
        const float v = mel[base + (size_t)t * N_MELS];
        M = A_COEF * M + S_COEF * v;
        out[base + (size_t)t * N_MELS] =
            sqrtf(v / powf(M + EPS_C, ALPHA_C) + 2.0f) - sqrt_delta;
    }
}

// ---------------------------------------------------------------------------
extern "C" void kernel_launch(void* const* d_in, const int* in_sizes, int n_in,
                              void* d_out, int out_size, void* d_ws, size_t ws_size,
                              hipStream_t stream)
{
    const float* x  = (const float*)d_in[0];   // (32, 160000) f32
    const float* fb = (const float*)d_in[1];   // (513, 128)   f32
    float* out = (float*)d_out;                // (32, 626, 128) f32

    float* spec = (float*)d_ws;                          // 20032 x 516 f32
    float* mel  = spec + (size_t)FRAMES_TOTAL * KP;      // 20032 x 128 f32
    float* E    = mel  + (size_t)FRAMES_TOTAL * N_MELS;  // 32 x 16 x 128 f32

    fft_mag_kernel<<<FRAMES_TOTAL, 256, 0, stream>>>(x, spec);
    mel_gemm_kernel<<<FRAMES_TOTAL / 16, 256, 0, stream>>>(spec, fb, mel);
    scan_chunk_kernel<<<BATCH * N_CHUNK, N_MELS, 0, stream>>>(mel, E);
    scan_combine_kernel<<<BATCH, N_MELS, 0, stream>>>(E);
    pcen_emit_kernel<<<BATCH * N_CHUNK, N_MELS, 0, stream>>>(mel, E, out);
}